// InteractionModel_4_52802327937169
// MI455X (gfx1250) — compile-verified
//
#include <hip/hip_runtime.h>
#include <math.h>

// ---------------------------------------------------------------------------
// InteractionModel on MI455X (gfx1250): fused flash-attention with fp32 WMMA
// + async global->LDS staging (ASYNCcnt) for load/compute overlap.
// ---------------------------------------------------------------------------

#define N_ROWS 8192
#define DIM    512
#define T_OUT  12

typedef __attribute__((ext_vector_type(2))) float v2f;
typedef __attribute__((ext_vector_type(8))) float v8f;
typedef __attribute__((ext_vector_type(4))) int   v4i;

#define GLOBAL_AS __attribute__((address_space(1)))
#define LDS_AS    __attribute__((address_space(3)))

// D = A(16x4) x B(4x16) + C, fp32 throughout (V_WMMA_F32_16X16X4_F32).
__device__ __forceinline__ v8f wmma_f32(v2f a, v2f b, v8f c) {
  return __builtin_amdgcn_wmma_f32_16x16x4_f32(false, a, false, b, (short)0, c,
                                               false, false);
}

// 16-byte global -> LDS copy. Prefers GLOBAL_LOAD_ASYNC_TO_LDS_B128 (no VGPR
// round-trip, tracked by ASYNCcnt); falls back to a synchronous float4 copy.
__device__ __forceinline__ void async_copy16(const float* gsrc, float* ldst) {
#if __has_builtin(__builtin_amdgcn_global_load_async_to_lds_b128)
  __builtin_amdgcn_global_load_async_to_lds_b128(
      (GLOBAL_AS v4i*)const_cast<float*>(gsrc), (LDS_AS v4i*)ldst, 0, 0);
#else
  const float4 v = *reinterpret_cast<const float4*>(gsrc);
  ldst[0] = v.x; ldst[1] = v.y; ldst[2] = v.z; ldst[3] = v.w;
#endif
}

__device__ __forceinline__ void wait_async() {
#if __has_builtin(__builtin_amdgcn_global_load_async_to_lds_b128)
#if __has_builtin(__builtin_amdgcn_s_wait_asynccnt)
  __builtin_amdgcn_s_wait_asynccnt(0);
#else
  asm volatile("s_wait_asynccnt 0x0" ::: "memory");
#endif
#endif
}

// ---------------------------------------------------------------------------
// LayerNorm: one row (512 floats) per 256-thread block.
// ---------------------------------------------------------------------------
__global__ void __launch_bounds__(256)
ln_kernel(const float* __restrict__ X, const float* __restrict__ g,
          const float* __restrict__ b, float* __restrict__ Y) {
  __shared__ float red[256];
  const int row = blockIdx.x, tid = threadIdx.x;
  const float2 v = reinterpret_cast<const float2*>(X + (size_t)row * DIM)[tid];
  red[tid] = v.x + v.y;
  __syncthreads();
  for (int o = 128; o > 0; o >>= 1) {
    if (tid < o) red[tid] += red[tid + o];
    __syncthreads();
  }
  const float mean = red[0] * (1.0f / DIM);
  __syncthreads();
  const float dx = v.x - mean, dy = v.y - mean;
  red[tid] = dx * dx + dy * dy;
  __syncthreads();
  for (int o = 128; o > 0; o >>= 1) {
    if (tid < o) red[tid] += red[tid + o];
    __syncthreads();
  }
  const float rs = rsqrtf(red[0] * (1.0f / DIM) + 1e-6f);
  const float2 gg = reinterpret_cast<const float2*>(g)[tid];
  const float2 bb = reinterpret_cast<const float2*>(b)[tid];
  float2 o2;
  o2.x = dx * rs * gg.x + bb.x;
  o2.y = dy * rs * gg.y + bb.y;
  reinterpret_cast<float2*>(Y + (size_t)row * DIM)[tid] = o2;
}

// ---------------------------------------------------------------------------
// Q projection: Qout = X @ W^T + bias   (M=8192, N=512, K=512), WMMA tiled,
// double-buffered async LDS staging.
// ---------------------------------------------------------------------------
constexpr int GK = 32;
constexpr int AS = GK + 4;  // LDS row stride (float4-aligned, conflict-free)

__global__ void __launch_bounds__(256)
qproj_kernel(const float* __restrict__ X, const float* __restrict__ W,
             const float* __restrict__ bias, float* __restrict__ Qout) {
  __shared__ float As[2][64 * AS];
  __shared__ float Bs[2][64 * AS];
  const int tid = threadIdx.x, lane = tid & 31, wave = tid >> 5;
  const int nl = lane & 15, klo = (lane >> 4) * 2;
  const int m0 = blockIdx.y * 64, n0 = blockIdx.x * 64;
  const int mb = wave >> 1, nb0 = (wave & 1) * 2;
  v8f c0, c1;
#pragma unroll
  for (int e = 0; e < 8; ++e) { c0[e] = 0.0f; c1[e] = 0.0f; }

  const int ar = mb * 16 + nl;    // A-frag row (M)
  const int br0 = nb0 * 16 + nl;  // B-frag row of W (N)

  // stage chunk k0 into buffer p (async issue, no wait)
  auto stage = [&](int p, int k0) {
    for (int idx = tid; idx < 64 * (GK / 4); idx += 256) {
      const int r = idx >> 3, c = (idx & 7) * 4;
      async_copy16(X + (size_t)(m0 + r) * DIM + k0 + c, &As[p][r * AS + c]);
      async_copy16(W + (size_t)(n0 + r) * DIM + k0 + c, &Bs[p][r * AS + c]);
    }
  };

  stage(0, 0);
  wait_async();
  __syncthreads();

  int p = 0;
  for (int k0 = 0; k0 < DIM; k0 += GK) {
    if (k0 + GK < DIM) stage(p ^ 1, k0 + GK);  // overlap next chunk load
    const float* Ap = As[p];
    const float* Bp = Bs[p];
#pragma unroll
    for (int kk = 0; kk < GK; kk += 4) {
      const v2f a  = *reinterpret_cast<const v2f*>(Ap + ar * AS + kk + klo);
      const v2f b0 = *reinterpret_cast<const v2f*>(Bp + br0 * AS + kk + klo);
      const v2f b1 = *reinterpret_cast<const v2f*>(Bp + (br0 + 16) * AS + kk + klo);
      c0 = wmma_f32(a, b0, c0);
      c1 = wmma_f32(a, b1, c1);
    }
    wait_async();
    __syncthreads();
    p ^= 1;
  }

  const int rb = mb * 16 + (lane >> 4) * 8;
  const int col0 = n0 + nb0 * 16 + nl;
#pragma unroll
  for (int e = 0; e < 8; ++e) {
    const size_t row = (size_t)(m0 + rb + e);
    Qout[row * DIM + col0]      = c0[e] + bias[col0];
    Qout[row * DIM + col0 + 16] = c1[e] + bias[col0 + 16];
  }
}

// ---------------------------------------------------------------------------
// Flash attention: F = (softmax(Q @ KV^T, axis=1) @ KV) / sqrt(D)
// Block = 256 threads (8 waves), BM=64 query rows, BN=64 key tile.
// Q block (64x512) pinned in LDS; KV tiles streamed in two 256-column halves
// so the second half's async load overlaps the first half of the S K-loop.
// ---------------------------------------------------------------------------
constexpr int BM = 64;
constexpr int BN = 64;
constexpr int QS = DIM + 2;  // LDS stride for Q / KV tiles (bank padding)
constexpr int SS = BN + 2;   // LDS stride for S/P tile

__global__ void __launch_bounds__(256)
attn_flash_kernel(const float* __restrict__ Q, const float* __restrict__ KV,
                  float* __restrict__ F) {
  extern __shared__ float smem[];
  float* Qs   = smem;                 // BM x QS
  float* Xs   = Qs + BM * QS;         // BN x QS
  float* Ps   = Xs + BN * QS;         // BM x SS (S, then P)
  float* mrow = Ps + BM * SS;         // running row max
  float* lrow = mrow + BM;            // running row sum
  float* crow = lrow + BM;            // per-tile correction

  const int tid  = threadIdx.x;
  const int lane = tid & 31;
  const int wave = tid >> 5;
  const int row0 = blockIdx.x * BM;
  const int nl   = lane & 15;
  const int klo  = (lane >> 4) * 2;

  // Stage the Q block once (async, reused for all 128 KV tiles).
  for (int idx = tid; idx < BM * (DIM / 4); idx += 256) {
    const int r = idx >> 7, c = (idx & 127) * 4;
    async_copy16(Q + (size_t)(row0 + r) * DIM + c, Qs + r * QS + c);
  }
  if (tid < BM) { mrow[tid] = -3.0e38f; lrow[tid] = 0.0f; }

  v8f acc[4][4];
#pragma unroll
  for (int i = 0; i < 4; ++i)
#pragma unroll
    for (int j = 0; j < 4; ++j)
#pragma unroll
      for (int e = 0; e < 8; ++e) acc[i][j][e] = 0.0f;

  wait_async();
  __syncthreads();

  // S-stage subtile assignment: wave w -> (mb=w>>1, nb=2*(w&1)+{0,1})
  const int s_mb  = wave >> 1;
  const int s_nb0 = (wave & 1) * 2;
  const int s_mr  = s_mb * 16 + nl;
  const int s_nr0 = s_nb0 * 16 + nl;

  for (int kt = 0; kt < N_ROWS / BN; ++kt) {
    const size_t kbase = (size_t)kt * BN * DIM;
    // stage columns [0,256)
    for (int idx = tid; idx < BN * 64; idx += 256) {
      const int r = idx >> 6, c = (idx & 63) * 4;
      async_copy16(KV + kbase + (size_t)r * DIM + c, Xs + r * QS + c);
    }
    wait_async();
    __syncthreads();
    // issue columns [256,512): overlaps with the first S K-half below
    for (int idx = tid; idx < BN * 64; idx += 256) {
      const int r = idx >> 6, c = (idx & 63) * 4;
      async_copy16(KV + kbase + (size_t)r * DIM + 256 + c,
                   Xs + r * QS + 256 + c);
    }

    // ---- S = Q_blk @ KV_tile^T : two 16x16 WMMA tiles per wave ----
    v8f s0, s1;
#pragma unroll
    for (int e = 0; e < 8; ++e) { s0[e] = 0.0f; s1[e] = 0.0f; }
    for (int k0 = 0; k0 < 256; k0 += 4) {
      const v2f a  = *reinterpret_cast<const v2f*>(Qs + s_mr * QS + k0 + klo);
      const v2f b0 = *reinterpret_cast<const v2f*>(Xs + s_nr0 * QS + k0 + klo);
      const v2f b1 =
          *reinterpret_cast<const v2f*>(Xs + (s_nr0 + 16) * QS + k0 + klo);
      s0 = wmma_f32(a, b0, s0);
      s1 = wmma_f32(a, b1, s1);
    }
    wait_async();
    __syncthreads();
    for (int k0 = 256; k0 < DIM; k0 += 4) {
      const v2f a  = *reinterpret_cast<const v2f*>(Qs + s_mr * QS + k0 + klo);
      const v2f b0 = *reinterpret_cast<const v2f*>(Xs + s_nr0 * QS + k0 + klo);
      const v2f b1 =
          *reinterpret_cast<const v2f*>(Xs + (s_nr0 + 16) * QS + k0 + klo);
      s0 = wmma_f32(a, b0, s0);
      s1 = wmma_f32(a, b1, s1);
    }
    {
      const int rb = s_mb * 16 + (lane >> 4) * 8;
      const int cc = s_nb0 * 16 + nl;
#pragma unroll
      for (int e = 0; e < 8; ++e) {
        Ps[(rb + e) * SS + cc]      = s0[e];
        Ps[(rb + e) * SS + cc + 16] = s1[e];
      }
    }
    __syncthreads();

    // ---- online softmax stats: one thread per query row ----
    if (tid < BM) {
      float* r = Ps + tid * SS;
      const float mo = mrow[tid];
      float mx = mo;
      for (int j = 0; j < BN; ++j) mx = fmaxf(mx, r[j]);
      const float c = __expf(mo - mx);
      float s = 0.0f;
      for (int j = 0; j < BN; ++j) {
        const float p = __expf(r[j] - mx);
        r[j] = p;
        s += p;
      }
      lrow[tid] = lrow[tid] * c + s;
      mrow[tid] = mx;
      crow[tid] = c;
    }
    __syncthreads();

    // ---- O = O*corr + P @ KV_tile (WMMA, wave owns 64-wide D-slab) ----
#pragma unroll
    for (int mb = 0; mb < 4; ++mb) {
      const int rb = mb * 16 + (lane >> 4) * 8;
      v8f c;
#pragma unroll
      for (int e = 0; e < 8; ++e) c[e] = crow[rb + e];
#pragma unroll
      for (int nb = 0; nb < 4; ++nb)
#pragma unroll
        for (int e = 0; e < 8; ++e) acc[mb][nb][e] *= c[e];
    }
#pragma unroll 4
    for (int k0 = 0; k0 < BN; k0 += 4) {
      v2f a[4], b[4];
#pragma unroll
      for (int mb = 0; mb < 4; ++mb)
        a[mb] = *reinterpret_cast<const v2f*>(Ps + (mb * 16 + nl) * SS + k0 + klo);
      const int kb = k0 + klo;
#pragma unroll
      for (int nb = 0; nb < 4; ++nb) {
        const int col = wave * 64 + nb * 16 + nl;
        v2f bb;
        bb.x = Xs[kb * QS + col];
        bb.y = Xs[(kb + 1) * QS + col];
        b[nb] = bb;
      }
#pragma unroll
      for (int mb = 0; mb < 4; ++mb)
#pragma unroll
        for (int nb = 0; nb < 4; ++nb)
          acc[mb][nb] = wmma_f32(a[mb], b[nb], acc[mb][nb]);
    }
    __syncthreads();
  }

  // epilogue: F = O / (l * sqrt(D))
  const float inv_sqrt_d = 0.04419417382415922f;  // 1/sqrt(512)
#pragma unroll
  for (int mb = 0; mb < 4; ++mb) {
    const int rb = mb * 16 + (lane >> 4) * 8;
#pragma unroll
    for (int e = 0; e < 8; ++e) {
      const float sc = inv_sqrt_d / lrow[rb + e];
      float* dst = F + (size_t)(row0 + rb + e) * DIM + wave * 64 + nl;
#pragma unroll
      for (int nb = 0; nb < 4; ++nb) dst[nb * 16] = acc[mb][nb][e] * sc;
    }
  }
}

// ---------------------------------------------------------------------------
// Gate: t2 = sigmoid((t @ Wt1^T + bt1) @ Wt2^T + bt2)  (scalar)
// ---------------------------------------------------------------------------
__global__ void gate_kernel(const float* __restrict__ t,
                            const float* __restrict__ Wt1,
                            const float* __restrict__ bt1,
                            const float* __restrict__ Wt2,
                            const float* __restrict__ bt2,
                            float* __restrict__ g) {
  if (threadIdx.x == 0) {
    const float tv = t[0];
    float s = bt2[0];
    for (int j = 0; j < 50; ++j) s += (tv * Wt1[j] + bt1[j]) * Wt2[j];
    g[0] = 1.0f / (1.0f + __expf(-s));
  }
}

// ---------------------------------------------------------------------------
// Head: out[i] = g*(relu([f1+x1; f2+x2]) @ Wp^T) + (1-g)*(relu([f3+x1; f4+x4]) @ Wp^T) + bp
// ---------------------------------------------------------------------------
__global__ void __launch_bounds__(256)
head_kernel(const float* __restrict__ f1, const float* __restrict__ f2,
            const float* __restrict__ f3, const float* __restrict__ f4,
            const float* __restrict__ x1, const float* __restrict__ x2,
            const float* __restrict__ x4, const float* __restrict__ Wp,
            const float* __restrict__ bp, const float* __restrict__ gate,
            float* __restrict__ out) {
  __shared__ float r12[T_OUT][256];
  __shared__ float r34[T_OUT][256];
  const int i = blockIdx.x, tid = threadIdx.x;
  float a12[T_OUT], a34[T_OUT];
#pragma unroll
  for (int t = 0; t < T_OUT; ++t) { a12[t] = 0.0f; a34[t] = 0.0f; }
  for (int k = tid; k < DIM; k += 256) {
    const size_t off = (size_t)i * DIM + k;
    const float h1 = fmaxf(f1[off] + x1[off], 0.0f);
    const float h2 = fmaxf(f2[off] + x2[off], 0.0f);
    const float h3 = fmaxf(f3[off] + x1[off], 0.0f);
    const float h4 = fmaxf(f4[off] + x4[off], 0.0f);
#pragma unroll
    for (int t = 0; t < T_OUT; ++t) {
      const float w1 = Wp[t * (2 * DIM) + k];
      const float w2 = Wp[t * (2 * DIM) + DIM + k];
      a12[t] += h1 * w1 + h2 * w2;
      a34[t] += h3 * w1 + h4 * w2;
    }
  }
#pragma unroll
  for (int t = 0; t < T_OUT; ++t) { r12[t][tid] = a12[t]; r34[t][tid] = a34[t]; }
  __syncthreads();
  for (int o = 128; o > 0; o >>= 1) {
    if (tid < o) {
#pragma unroll
      for (int t = 0; t < T_OUT; ++t) {
        r12[t][tid] += r12[t][tid + o];
        r34[t][tid] += r34[t][tid + o];
      }
    }
    __syncthreads();
  }
  if (tid < T_OUT) {
    const float g = gate[0];
    out[(size_t)i * T_OUT + tid] =
        g * r12[tid][0] + (1.0f - g) * r34[tid][0] + bp[tid];
  }
}

// ---------------------------------------------------------------------------
extern "C" void kernel_launch(void* const* d_in, const int* in_sizes, int n_in,
                              void* d_out, int out_size, void* d_ws,
                              size_t ws_size, hipStream_t stream) {
  (void)in_sizes; (void)n_in; (void)out_size; (void)ws_size;
  const float* xp   = (const float*)d_in[0];
  const float* xm   = (const float*)d_in[1];
  const float* xs   = (const float*)d_in[2];
  const float* ln_g = (const float*)d_in[3];
  const float* ln_b = (const float*)d_in[4];
  const float* W1   = (const float*)d_in[5];
  const float* b1   = (const float*)d_in[6];
  const float* W2   = (const float*)d_in[7];
  const float* b2   = (const float*)d_in[8];
  const float* W3   = (const float*)d_in[9];
  const float* b3   = (const float*)d_in[10];
  const float* W4   = (const float*)d_in[11];
  const float* b4   = (const float*)d_in[12];
  const float* Wp   = (const float*)d_in[13];
  const float* bp   = (const float*)d_in[14];
  const float* tt   = (const float*)d_in[15];
  const float* Wt1  = (const float*)d_in[16];
  const float* bt1  = (const float*)d_in[17];
  const float* Wt2  = (const float*)d_in[18];
  const float* bt2  = (const float*)d_in[19];
  float* out = (float*)d_out;

  float* ws = (float*)d_ws;
  const size_t ND = (size_t)N_ROWS * DIM;
  float* x1  = ws;
  float* x2  = x1 + ND;
  float* x4  = x2 + ND;
  float* q12 = x4 + ND;
  float* q21 = q12 + ND;
  float* q34 = q21 + ND;
  float* q43 = q34 + ND;
  // Attention outputs alias their (fully consumed, per-block) Q buffers.
  float* f2 = q12;
  float* f1 = q21;
  float* f4 = q34;
  float* f3 = q43;
  float* gate = q43 + ND;

  ln_kernel<<<N_ROWS, 256, 0, stream>>>(xp, ln_g, ln_b, x1);
  ln_kernel<<<N_ROWS, 256, 0, stream>>>(xm, ln_g, ln_b, x2);
  ln_kernel<<<N_ROWS, 256, 0, stream>>>(xs, ln_g, ln_b, x4);

  dim3 gq(DIM / 64, N_ROWS / 64);
  qproj_kernel<<<gq, 256, 0, stream>>>(x1, W1, b1, q12);
  qproj_kernel<<<gq, 256, 0, stream>>>(x2, W2, b2, q21);
  qproj_kernel<<<gq, 256, 0, stream>>>(x1, W3, b3, q34);
  qproj_kernel<<<gq, 256, 0, stream>>>(x4, W4, b4, q43);

  const size_t smem =
      (size_t)(BM * QS + BN * QS + BM * SS + 3 * BM) * sizeof(float);
  static bool attr_set = [] {
    (void)hipFuncSetAttribute(reinterpret_cast<const void*>(attn_flash_kernel),
                              hipFuncAttributeMaxDynamicSharedMemorySize,
                              (int)((size_t)(BM * QS + BN * QS + BM * SS + 3 * BM) *
                                    sizeof(float)));
    return true;
  }();
  (void)attr_set;
  attn_flash_kernel<<<N_ROWS / BM, 256, smem, stream>>>(q12, x2, f2);
  attn_flash_kernel<<<N_ROWS / BM, 256, smem, stream>>>(q21, x1, f1);
  attn_flash_kernel<<<N_ROWS / BM, 256, smem, stream>>>(q34, x4, f4);
  attn_flash_kernel<<<N_ROWS / BM, 256, smem, stream>>>(q43, x1, f3);

  gate_kernel<<<1, 32, 0, stream>>>(tt, Wt1, bt1, Wt2, bt2, gate);
  head_kernel<<<N_ROWS, 256, 0, stream>>>(f1, f2, f3, f4, x1, x2, x4, Wp, bp,
                                          gate, out);
}